// RnnEntityModel_14388140442182
// MI455X (gfx1250) — compile-verified
//
#include <hip/hip_runtime.h>
#include <hip/hip_bf16.h>
#include <math.h>

// ---------------------------------------------------------------------------
// Bi-GRU entity model on gfx1250 (MI455X), fp32 via V_WMMA_F32_16X16X4_F32.
// Weights are pre-packed into WMMA-fragment order so every B-fragment load is
// one fully coalesced 256B wave-load (the recurrence is weight-BW bound).
// ---------------------------------------------------------------------------

#define Bn    64
#define Tn    512
#define Hn    1000
#define NCn   2000
#define FEATn 500
#define FEATP 512          // padded feature stride (8B-aligned float2 rows)
#define G3    3000         // 3*H
#define XPS   3008         // padded gate-row stride (188 tiles * 16)
#define NT3   188          // ceil(3000/16)
#define NTC   125          // 2000/16
#define EPSn  1e-5f

typedef __attribute__((ext_vector_type(2))) float v2f;
typedef __attribute__((ext_vector_type(8))) float v8f;

// ---------------------------------------------------------------------------
// 0) Pack a row-major [nrows x K] weight (out = in @ W^T) into WMMA B-fragment
//    order: dst[((nt*(K/4) + ks)*32 + lane)] = float2{ W[n][kb], W[n][kb+1] }
//    with n = nt*16 + (lane&15) (clamped), kb = 4*ks + 2*(lane>>4).
// ---------------------------------------------------------------------------
__global__ void pack_weights_kernel(const float* __restrict__ src, float* __restrict__ dst,
                                    int nrows, int K, int ntiles) {
  const int idx = blockIdx.x * blockDim.x + threadIdx.x;
  const int total = ntiles * (K / 4) * 32;
  if (idx >= total) return;
  const int lane = idx & 31;
  const int rest = idx >> 5;
  const int kq   = K >> 2;
  const int ks   = rest % kq;
  const int nt   = rest / kq;
  int n = nt * 16 + (lane & 15);
  if (n >= nrows) n = nrows - 1;                 // tail-tile clamp (store-masked later)
  const int kb = 4 * ks + 2 * (lane >> 4);
  ((v2f*)dst)[idx] = *(const v2f*)(src + (size_t)n * K + kb);
}

// ---------------------------------------------------------------------------
// 1) Embedding gather + concat -> feat[B][T][FEATP] (fp32, zero-padded tail)
// ---------------------------------------------------------------------------
__global__ void embed_concat_kernel(const int* __restrict__ wi, const int* __restrict__ pi,
                                    const int* __restrict__ ii, const int* __restrict__ ti,
                                    const int* __restrict__ mi,
                                    const float* __restrict__ we, const float* __restrict__ pe,
                                    const float* __restrict__ ie, const float* __restrict__ te,
                                    const float* __restrict__ me,
                                    float* __restrict__ feat) {
  const int t = blockIdx.x;
  const int b = blockIdx.y;
  const int s = b * Tn + t;
  const int w  = wi[s];
  const int p  = pi[s];
  const int io = ii[s];
  const int ty = ti[s];
  const int m  = mi[s];
  float* o = feat + (size_t)s * FEATP;
  for (int f = threadIdx.x; f < FEATP; f += blockDim.x) {
    float v;
    if      (f < 300) v = we[(size_t)w * 300 + f];
    else if (f < 350) v = pe[p  * 50 + (f - 300)];
    else if (f < 400) v = ie[io * 50 + (f - 350)];
    else if (f < 450) v = te[ty * 50 + (f - 400)];
    else if (f < 500) v = me[m  * 50 + (f - 450)];
    else              v = 0.0f;
    o[f] = v;
  }
}

// ---------------------------------------------------------------------------
// 2) Per-timestep projection GEMMs (both directions), f32 WMMA 16x16x4.
//    Each wave: one 16-wide N stripe, all 4 M-tiles (M = B = 64), so each
//    packed weight fragment feeds 4 WMMAs.
//    grid: (47, 2) blocks x 128 threads (4 waves) -> 188 N-tiles x 2 dirs.
// ---------------------------------------------------------------------------
__global__ void gru_step_gemm_kernel(const float* __restrict__ feat,
                                     const float* __restrict__ hst,
                                     const float* __restrict__ wih_pack,   // [2][NT3*16*FEATn]
                                     const float* __restrict__ whh_pack,   // [2][NT3*16*Hn]
                                     const int* __restrict__ len,
                                     float* __restrict__ xp, float* __restrict__ hp,
                                     int t) {
  const int lane = threadIdx.x & 31;
  const int wave = threadIdx.x >> 5;
  const int dir  = blockIdx.y;
  const int nt   = blockIdx.x * 4 + wave;     // 0..187
  const int l15  = lane & 15;
  const int half = lane >> 4;                 // selects K pair {k,k+1} / {k+2,k+3}
  const int n    = nt * 16 + l15;

  // Per-lane A-row pointers for the input projection (handles bwd reversal).
  const float* arow0; const float* arow1; const float* arow2; const float* arow3;
  {
    const float* ar[4];
#pragma unroll
    for (int mt = 0; mt < 4; ++mt) {
      const int b = mt * 16 + l15;
      int tt = t;
      if (dir) { tt = len[b] - 1 - t; if (tt < 0) tt = 0; }
      ar[mt] = feat + ((size_t)b * Tn + tt) * FEATP;
    }
    arow0 = ar[0]; arow1 = ar[1]; arow2 = ar[2]; arow3 = ar[3];
  }

  // ---- input projection: K = FEAT = 500 ----
  v8f ax0 = {}, ax1 = {}, ax2 = {}, ax3 = {};
  {
    const v2f* bp = (const v2f*)(wih_pack + (size_t)dir * NT3 * 16 * FEATn)
                    + (size_t)nt * (FEATn / 4) * 32 + lane;
    for (int i = 0; i < FEATn / 4; ++i) {
      const v2f bf = bp[(size_t)i * 32];      // coalesced 256B wave-load
      const int kb = 4 * i + 2 * half;
      const v2f a0 = *(const v2f*)(arow0 + kb);
      const v2f a1 = *(const v2f*)(arow1 + kb);
      const v2f a2 = *(const v2f*)(arow2 + kb);
      const v2f a3 = *(const v2f*)(arow3 + kb);
      ax0 = __builtin_amdgcn_wmma_f32_16x16x4_f32(false, a0, false, bf, (short)0, ax0, false, false);
      ax1 = __builtin_amdgcn_wmma_f32_16x16x4_f32(false, a1, false, bf, (short)0, ax1, false, false);
      ax2 = __builtin_amdgcn_wmma_f32_16x16x4_f32(false, a2, false, bf, (short)0, ax2, false, false);
      ax3 = __builtin_amdgcn_wmma_f32_16x16x4_f32(false, a3, false, bf, (short)0, ax3, false, false);
    }
  }
  // Store xp now to release accumulator registers.
  if (n < G3) {
    float* o = xp + (size_t)dir * Bn * XPS;
#pragma unroll
    for (int i = 0; i < 8; ++i) {
      const int r = i + 8 * half;
      o[(size_t)(r +  0) * XPS + n] = ax0[i];
      o[(size_t)(r + 16) * XPS + n] = ax1[i];
      o[(size_t)(r + 32) * XPS + n] = ax2[i];
      o[(size_t)(r + 48) * XPS + n] = ax3[i];
    }
  }

  // ---- hidden projection: K = H = 1000 ----
  v8f ah0 = {}, ah1 = {}, ah2 = {}, ah3 = {};
  {
    const float* hb = hst + (size_t)dir * Bn * Hn;
    const float* h0 = hb + (size_t)(0  + l15) * Hn;
    const float* h1 = hb + (size_t)(16 + l15) * Hn;
    const float* h2 = hb + (size_t)(32 + l15) * Hn;
    const float* h3 = hb + (size_t)(48 + l15) * Hn;
    const v2f* bp = (const v2f*)(whh_pack + (size_t)dir * NT3 * 16 * Hn)
                    + (size_t)nt * (Hn / 4) * 32 + lane;
    for (int i = 0; i < Hn / 4; ++i) {
      const v2f bf = bp[(size_t)i * 32];
      const int kb = 4 * i + 2 * half;
      const v2f a0 = *(const v2f*)(h0 + kb);
      const v2f a1 = *(const v2f*)(h1 + kb);
      const v2f a2 = *(const v2f*)(h2 + kb);
      const v2f a3 = *(const v2f*)(h3 + kb);
      ah0 = __builtin_amdgcn_wmma_f32_16x16x4_f32(false, a0, false, bf, (short)0, ah0, false, false);
      ah1 = __builtin_amdgcn_wmma_f32_16x16x4_f32(false, a1, false, bf, (short)0, ah1, false, false);
      ah2 = __builtin_amdgcn_wmma_f32_16x16x4_f32(false, a2, false, bf, (short)0, ah2, false, false);
      ah3 = __builtin_amdgcn_wmma_f32_16x16x4_f32(false, a3, false, bf, (short)0, ah3, false, false);
    }
  }
  if (n < G3) {
    float* o = hp + (size_t)dir * Bn * XPS;
#pragma unroll
    for (int i = 0; i < 8; ++i) {
      const int r = i + 8 * half;
      o[(size_t)(r +  0) * XPS + n] = ah0[i];
      o[(size_t)(r + 16) * XPS + n] = ah1[i];
      o[(size_t)(r + 32) * XPS + n] = ah2[i];
      o[(size_t)(r + 48) * XPS + n] = ah3[i];
    }
  }
}

// ---------------------------------------------------------------------------
// 3) GRU gate update (elementwise), masked by sequence length.
// ---------------------------------------------------------------------------
__global__ void gru_gate_kernel(const float* __restrict__ xp, const float* __restrict__ hp,
                                const float* __restrict__ bih_f, const float* __restrict__ bhh_f,
                                const float* __restrict__ bih_b, const float* __restrict__ bhh_b,
                                const int* __restrict__ len,
                                float* __restrict__ hst, int t) {
  const int idx = blockIdx.x * blockDim.x + threadIdx.x;   // [0, 2*B*H)
  if (idx >= 2 * Bn * Hn) return;
  const int dir = idx / (Bn * Hn);
  const int r0  = idx - dir * Bn * Hn;
  const int b   = r0 / Hn;
  const int j   = r0 - b * Hn;
  if (t >= len[b]) return;                                 // freeze past seq end

  const float* bih = dir ? bih_b : bih_f;
  const float* bhh = dir ? bhh_b : bhh_f;
  const float* xrw = xp + (size_t)dir * Bn * XPS + (size_t)b * XPS;
  const float* hrw = hp + (size_t)dir * Bn * XPS + (size_t)b * XPS;

  const float xr = xrw[j]          + bih[j];
  const float xz = xrw[Hn + j]     + bih[Hn + j];
  const float xn = xrw[2 * Hn + j] + bih[2 * Hn + j];
  const float hr = hrw[j]          + bhh[j];
  const float hz = hrw[Hn + j]     + bhh[Hn + j];
  const float hn = hrw[2 * Hn + j] + bhh[2 * Hn + j];

  const float r  = 1.0f / (1.0f + expf(-(xr + hr)));
  const float z  = 1.0f / (1.0f + expf(-(xz + hz)));
  const float nn = tanhf(xn + r * hn);
  const float hprev = hst[idx];
  hst[idx] = (1.0f - z) * nn + z * hprev;
}

// ---------------------------------------------------------------------------
// 4) Final linear [64,2000] = h_cat[64,2000] @ lin_w.T + bias, fused BN.
//    grid: 32 blocks x 128 threads (4 waves) -> 125 N-tiles (last 3 waves idle)
// ---------------------------------------------------------------------------
__global__ void final_linear_bn_kernel(const float* __restrict__ hst,
                                       const float* __restrict__ lw_pack,
                                       const float* __restrict__ lb,
                                       const float* __restrict__ gam, const float* __restrict__ bet,
                                       const float* __restrict__ mea, const float* __restrict__ var,
                                       float* __restrict__ out) {
  const int lane = threadIdx.x & 31;
  const int wave = threadIdx.x >> 5;
  const int nt   = blockIdx.x * 4 + wave;
  if (nt >= NTC) return;                      // wave-uniform exit (EXEC stays all-1)
  const int l15  = lane & 15;
  const int half = lane >> 4;
  const int n    = nt * 16 + l15;             // < 2000 always

  v8f a0 = {}, a1 = {}, a2 = {}, a3 = {};
  {
    const v2f* bp = (const v2f*)lw_pack + (size_t)nt * ((2 * Hn) / 4) * 32 + lane;
    for (int i = 0; i < (2 * Hn) / 4; ++i) {
      const v2f bf = bp[(size_t)i * 32];
      const int kb = 4 * i + 2 * half;        // pair never straddles H boundary
      const size_t koff = (kb < Hn) ? (size_t)kb : ((size_t)Bn * Hn + (kb - Hn));
      const v2f f0 = *(const v2f*)(hst + (size_t)(0  + l15) * Hn + koff);
      const v2f f1 = *(const v2f*)(hst + (size_t)(16 + l15) * Hn + koff);
      const v2f f2 = *(const v2f*)(hst + (size_t)(32 + l15) * Hn + koff);
      const v2f f3 = *(const v2f*)(hst + (size_t)(48 + l15) * Hn + koff);
      a0 = __builtin_amdgcn_wmma_f32_16x16x4_f32(false, f0, false, bf, (short)0, a0, false, false);
      a1 = __builtin_amdgcn_wmma_f32_16x16x4_f32(false, f1, false, bf, (short)0, a1, false, false);
      a2 = __builtin_amdgcn_wmma_f32_16x16x4_f32(false, f2, false, bf, (short)0, a2, false, false);
      a3 = __builtin_amdgcn_wmma_f32_16x16x4_f32(false, f3, false, bf, (short)0, a3, false, false);
    }
  }
  const float g  = gam[n];
  const float bt = bet[n];
  const float mn = mea[n];
  const float iv = rsqrtf(var[n] + EPSn);
  const float lbn = lb[n];
#pragma unroll
  for (int i = 0; i < 8; ++i) {
    const int r = i + 8 * half;
    out[(size_t)(r +  0) * NCn + n] = g * ((a0[i] + lbn) - mn) * iv + bt;
    out[(size_t)(r + 16) * NCn + n] = g * ((a1[i] + lbn) - mn) * iv + bt;
    out[(size_t)(r + 32) * NCn + n] = g * ((a2[i] + lbn) - mn) * iv + bt;
    out[(size_t)(r + 48) * NCn + n] = g * ((a3[i] + lbn) - mn) * iv + bt;
  }
}

// ---------------------------------------------------------------------------
// Host-side orchestration
// ---------------------------------------------------------------------------
extern "C" void kernel_launch(void* const* d_in, const int* in_sizes, int n_in,
                              void* d_out, int out_size, void* d_ws, size_t ws_size,
                              hipStream_t stream) {
  (void)in_sizes; (void)n_in; (void)out_size; (void)ws_size;

  const int*   word_idxs    = (const int*)  d_in[0];
  const int*   pos_idxs     = (const int*)  d_in[1];
  const int*   iob_idxs     = (const int*)  d_in[2];
  const int*   type_idxs    = (const int*)  d_in[3];
  const int*   mention_idxs = (const int*)  d_in[4];
  const int*   lengths      = (const int*)  d_in[5];
  const float* hidden       = (const float*)d_in[6];
  const float* word_emb     = (const float*)d_in[7];
  const float* pos_emb      = (const float*)d_in[8];
  const float* iob_emb      = (const float*)d_in[9];
  const float* type_emb     = (const float*)d_in[10];
  const float* mention_emb  = (const float*)d_in[11];
  const float* w_ih_fwd     = (const float*)d_in[12];
  const float* w_hh_fwd     = (const float*)d_in[13];
  const float* b_ih_fwd     = (const float*)d_in[14];
  const float* b_hh_fwd     = (const float*)d_in[15];
  const float* w_ih_bwd     = (const float*)d_in[16];
  const float* w_hh_bwd     = (const float*)d_in[17];
  const float* b_ih_bwd     = (const float*)d_in[18];
  const float* b_hh_bwd     = (const float*)d_in[19];
  const float* lin_w        = (const float*)d_in[20];
  const float* lin_b        = (const float*)d_in[21];
  const float* bn_gamma     = (const float*)d_in[22];
  const float* bn_beta      = (const float*)d_in[23];
  const float* bn_mean      = (const float*)d_in[24];
  const float* bn_var       = (const float*)d_in[25];

  // workspace layout (floats), total ~123 MB
  float* ws       = (float*)d_ws;
  float* feat     = ws;                                        // B*T*FEATP = 16,777,216
  float* xp       = feat     + (size_t)Bn * Tn * FEATP;        // 2*B*XPS   =    385,024
  float* hp       = xp       + (size_t)2 * Bn * XPS;           // 2*B*XPS   =    385,024
  float* hst      = hp       + (size_t)2 * Bn * XPS;           // 2*B*H     =    128,000
  float* wih_pack = hst      + (size_t)2 * Bn * Hn;            // 2*3008*500 = 3,008,000
  float* whh_pack = wih_pack + (size_t)2 * NT3 * 16 * FEATn;   // 2*3008*1000= 6,016,000
  float* lw_pack  = whh_pack + (size_t)2 * NT3 * 16 * Hn;      // 2000*2000  = 4,000,000

  // 0) pack weights into WMMA B-fragment order (coalesced inner-loop loads)
  {
    const int tot_ih = NT3 * (FEATn / 4) * 32;                 // 752,000
    const int tot_hh = NT3 * (Hn / 4) * 32;                    // 1,504,000
    const int tot_lw = NTC * ((2 * Hn) / 4) * 32;              // 2,000,000
    pack_weights_kernel<<<(tot_ih + 255) / 256, 256, 0, stream>>>(
        w_ih_fwd, wih_pack, G3, FEATn, NT3);
    pack_weights_kernel<<<(tot_ih + 255) / 256, 256, 0, stream>>>(
        w_ih_bwd, wih_pack + (size_t)NT3 * 16 * FEATn, G3, FEATn, NT3);
    pack_weights_kernel<<<(tot_hh + 255) / 256, 256, 0, stream>>>(
        w_hh_fwd, whh_pack, G3, Hn, NT3);
    pack_weights_kernel<<<(tot_hh + 255) / 256, 256, 0, stream>>>(
        w_hh_bwd, whh_pack + (size_t)NT3 * 16 * Hn, G3, Hn, NT3);
    pack_weights_kernel<<<(tot_lw + 255) / 256, 256, 0, stream>>>(
        lin_w, lw_pack, NCn, 2 * Hn, NTC);
  }

  // 1) embeddings -> feat
  embed_concat_kernel<<<dim3(Tn, Bn), 128, 0, stream>>>(
      word_idxs, pos_idxs, iob_idxs, type_idxs, mention_idxs,
      word_emb, pos_emb, iob_emb, type_emb, mention_emb, feat);

  // 2) h0 <- hidden
  hipMemcpyAsync(hst, hidden, (size_t)2 * Bn * Hn * sizeof(float),
                 hipMemcpyDeviceToDevice, stream);

  // 3) recurrence: per-step fused projection GEMMs + gate update
  for (int t = 0; t < Tn; ++t) {
    gru_step_gemm_kernel<<<dim3(47, 2), 128, 0, stream>>>(
        feat, hst, wih_pack, whh_pack, lengths, xp, hp, t);
    gru_gate_kernel<<<(2 * Bn * Hn + 255) / 256, 256, 0, stream>>>(
        xp, hp, b_ih_fwd, b_hh_fwd, b_ih_bwd, b_hh_bwd, lengths, hst, t);
  }

  // 4) final linear + batchnorm -> d_out[0 : B*NC)
  final_linear_bn_kernel<<<32, 128, 0, stream>>>(
      hst, lw_pack, lin_b, bn_gamma, bn_beta, bn_mean, bn_var, (float*)d_out);

  // 5) hidden_out [2,B,H] -> d_out[B*NC : B*NC + 2*B*H)
  hipMemcpyAsync((float*)d_out + (size_t)Bn * NCn, hst,
                 (size_t)2 * Bn * Hn * sizeof(float),
                 hipMemcpyDeviceToDevice, stream);
}